// ChebBen2BN_71159018160657
// MI455X (gfx1250) — compile-verified
//
#include <hip/hip_runtime.h>

typedef __attribute__((ext_vector_type(2))) float v2f;
typedef __attribute__((ext_vector_type(8))) float v8f;

#define N_NODES 100000
#define E_EDGES 1600000
#define DIN_C   128     // feature dim entering both GEMMs (x and h are 128-wide)
#define HID_C   128
#define OUT_C   64
#define BN_EPS  1e-5f

// ---------------------------------------------------------------- utilities
__global__ void zero_f32(float* __restrict__ p, long n) {
    long i = (long)blockIdx.x * blockDim.x + threadIdx.x;
    if (i < n) p[i] = 0.0f;
}

__global__ void deg_kernel(const int* __restrict__ src, float* __restrict__ deg, int e) {
    int i = blockIdx.x * blockDim.x + threadIdx.x;
    if (i < e) atomicAdd(&deg[src[i]], 1.0f);
}

__global__ void dis_kernel(float* __restrict__ deg_to_dis, int n) {
    int i = blockIdx.x * blockDim.x + threadIdx.x;
    if (i < n) {
        float d = deg_to_dis[i];
        deg_to_dis[i] = (d > 0.0f) ? rsqrtf(fmaxf(d, 1.0f)) : 0.0f;
    }
}

__global__ void norm_kernel(const int* __restrict__ src, const int* __restrict__ dst,
                            const float* __restrict__ dis, float* __restrict__ norm, int e) {
    int i = blockIdx.x * blockDim.x + threadIdx.x;
    if (i < e) norm[i] = -dis[src[i]] * dis[dst[i]];
}

// y[dst[e], c..c+3] += norm[e] * x[src[e], c..c+3]; one wave per edge
// (ed = tid>>5 is wave-uniform -> uniform index loads, 512B coalesced features)
__global__ void propagate_kernel(const int* __restrict__ src, const int* __restrict__ dst,
                                 const float* __restrict__ norm,
                                 const float* __restrict__ X, float* __restrict__ Y, long e) {
    long tid = (long)blockIdx.x * blockDim.x + threadIdx.x;
    long tot = e * (DIN_C / 4);
    if (tid >= tot) return;
    long ed = tid >> 5;                 // DIN_C/4 == 32 chunks per edge
    int  c4 = (int)(tid & 31) << 2;
    int  s = src[ed], d = dst[ed];
    float w = norm[ed];
    const float4 v = *(const float4*)(X + (size_t)s * DIN_C + c4);
    float* yp = Y + (size_t)d * DIN_C + c4;
    atomicAdd(yp + 0, w * v.x);
    atomicAdd(yp + 1, w * v.y);
    atomicAdd(yp + 2, w * v.z);
    atomicAdd(yp + 3, w * v.w);
}

// T2 = 2*T2 - T0   (Chebyshev recurrence finalize)
__global__ void cheb_finalize(float* __restrict__ T2, const float* __restrict__ T0, long n) {
    long i = (long)blockIdx.x * blockDim.x + threadIdx.x;
    if (i < n) T2[i] = 2.0f * T2[i] - T0[i];
}

// ------------------------------------------------------- fused GEMM + BN(+ReLU)
// Y[m, n] = BN( sum_k (A_k @ W[k])[m, n] + bias[n] )
// Block: 256 threads = 8 waves; 8 row-tiles x one 32-col tile. Each wave owns a
// 16x32 tile: two independent v8f accumulator chains (XDL pipelining), one A
// fetch feeds two WMMAs (halves A-stream bandwidth vs 16x16 tiles).
//
// LDS W slab stored K-PAIR-INTERLEAVED: Wl[k][row>>1][col][row&1], so each
// B operand (two consecutive K values for one column) is one aligned
// ds_load_b64 directly into the WMMA source VGPR pair -> no v_mov re-pairing.
// Slab = 3 x 128 x 32 fp32 = 48 KB (>=6 blocks resident per 320KB WGP).
template <int COUT, bool RELU>
__global__ __launch_bounds__(256)
void cheb_gemm_bn(const float* __restrict__ A0, const float* __restrict__ A1,
                  const float* __restrict__ A2, const float* __restrict__ W,
                  const float* __restrict__ bias,
                  const float* __restrict__ g, const float* __restrict__ beta,
                  const float* __restrict__ mean, const float* __restrict__ var,
                  float* __restrict__ Y, int nrows) {
    __shared__ float Wl[3 * DIN_C * 32];

    const int n0 = blockIdx.y * 32;
    // cooperative LDS fill: element (k, r, c) -> pair-interleaved slot
    //   Wl[(k*64 + (r>>1))*64 + c*2 + (r&1)]
    for (int i = threadIdx.x; i < 3 * DIN_C * 32; i += 256) {
        int k = i / (DIN_C * 32);
        int r = (i >> 5) & (DIN_C - 1);
        int c = i & 31;
        Wl[(k * 64 + (r >> 1)) * 64 + c * 2 + (r & 1)] =
            W[(size_t)k * DIN_C * COUT + (size_t)r * COUT + n0 + c];
    }
    __syncthreads();

    const int lane = threadIdx.x & 31;
    const int wave = threadIdx.x >> 5;
    const int m0 = (blockIdx.x * 8 + wave) * 16;
    if (m0 >= nrows) return;   // wave-uniform: EXEC stays all-ones for WMMA

    // V_WMMA_F32_16X16X4_F32 operand mapping (ISA 7.12.2):
    //   A 16x4: lanes 0-15 -> M=lane, K={0,1}; lanes 16-31 -> M=lane-16, K={2,3}
    //   B 4x16: mirrored (VGPR0=K kbase, VGPR1=K kbase+1; N = lane&15)
    //   C/D:    VGPR v -> M = v + 8*(lane>=16), N = lane&15
    const int row   = lane & 15;
    const int hi    = lane >> 4;          // 0: K pair {4t,4t+1}; 1: {4t+2,4t+3}
    const int kbase = hi << 1;
    const int col   = lane & 15;

    v8f acc0 = {0.f, 0.f, 0.f, 0.f, 0.f, 0.f, 0.f, 0.f};
    v8f acc1 = {0.f, 0.f, 0.f, 0.f, 0.f, 0.f, 0.f, 0.f};

    const float* Amats[3] = {A0, A1, A2};
#pragma unroll
    for (int k = 0; k < 3; ++k) {
        const float* Ap = Amats[k] + (size_t)(m0 + row) * DIN_C + kbase;
        // pair index p = k*64 + 2t + hi; addr = p*64 + col*2 (+32 for cols 16..31)
        const float* Bp = Wl + (k * 64 + hi) * 64 + col * 2;
#pragma unroll 4
        for (int t = 0; t < DIN_C / 4; ++t) {
            v2f a  = *(const v2f*)(Ap + 4 * t);          // global_load_b64
            v2f b0 = *(const v2f*)(Bp + t * 128);        // ds_load_b64
            v2f b1 = *(const v2f*)(Bp + t * 128 + 32);   // ds_load_b64
            acc0 = __builtin_amdgcn_wmma_f32_16x16x4_f32(
                false, a, false, b0, (short)0, acc0, false, false);
            acc1 = __builtin_amdgcn_wmma_f32_16x16x4_f32(
                false, a, false, b1, (short)0, acc1, false, false);
        }
    }

    // fused bias + eval-mode BN (+ ReLU): (acc + bias - mean)*g*rsqrt(var+eps) + beta
    const int mofs = hi << 3;
#pragma unroll
    for (int half = 0; half < 2; ++half) {
        const int n = n0 + 16 * half + col;
        const float sc = g[n] * rsqrtf(var[n] + BN_EPS);
        const float sh = beta[n] + (bias[n] - mean[n]) * sc;
#pragma unroll
        for (int v = 0; v < 8; ++v) {
            float val = (half ? acc1[v] : acc0[v]) * sc + sh;
            if (RELU) val = fmaxf(val, 0.0f);
            Y[(size_t)(m0 + v + mofs) * COUT + n] = val;
        }
    }
}

// ---------------------------------------------------------------- launcher
static inline unsigned cdiv(long a, long b) { return (unsigned)((a + b - 1) / b); }

extern "C" void kernel_launch(void* const* d_in, const int* in_sizes, int n_in,
                              void* d_out, int out_size, void* d_ws, size_t ws_size,
                              hipStream_t stream) {
    const float* x     = (const float*)d_in[0];
    const int*   ei    = (const int*)d_in[1];
    const float* W1    = (const float*)d_in[2];
    const float* b1    = (const float*)d_in[3];
    const float* W2    = (const float*)d_in[4];
    const float* b2    = (const float*)d_in[5];
    const float* g1    = (const float*)d_in[6];
    const float* beta1 = (const float*)d_in[7];
    const float* m1    = (const float*)d_in[8];
    const float* v1    = (const float*)d_in[9];
    const float* g2    = (const float*)d_in[10];
    const float* beta2 = (const float*)d_in[11];
    const float* m2    = (const float*)d_in[12];
    const float* v2    = (const float*)d_in[13];
    float* out = (float*)d_out;

    const int* src = ei;
    const int* dst = ei + E_EDGES;

    // workspace carve-out
    float* dis  = (float*)d_ws;                       // N
    float* norm = dis + N_NODES;                      // E
    float* T1   = norm + E_EDGES;                     // N*128
    float* T2   = T1 + (size_t)N_NODES * DIN_C;       // N*128
    float* h    = T2 + (size_t)N_NODES * DIN_C;       // N*128

    const long feat_elems = (long)N_NODES * DIN_C;
    const long prop_threads = (long)E_EDGES * (DIN_C / 4);
    const unsigned B = 256;

    // degree -> dis -> norm
    zero_f32<<<cdiv(N_NODES, B), B, 0, stream>>>(dis, N_NODES);
    deg_kernel<<<cdiv(E_EDGES, B), B, 0, stream>>>(src, dis, E_EDGES);
    dis_kernel<<<cdiv(N_NODES, B), B, 0, stream>>>(dis, N_NODES);
    norm_kernel<<<cdiv(E_EDGES, B), B, 0, stream>>>(src, dst, dis, norm, E_EDGES);

    const unsigned gx = cdiv((N_NODES + 15) / 16, 8);   // row-tile blocks

    // ---- layer 1: T0 = x, T1 = L x, T2 = 2 L T1 - x
    zero_f32<<<cdiv(feat_elems, B), B, 0, stream>>>(T1, feat_elems);
    propagate_kernel<<<cdiv(prop_threads, B), B, 0, stream>>>(src, dst, norm, x, T1, E_EDGES);
    zero_f32<<<cdiv(feat_elems, B), B, 0, stream>>>(T2, feat_elems);
    propagate_kernel<<<cdiv(prop_threads, B), B, 0, stream>>>(src, dst, norm, T1, T2, E_EDGES);
    cheb_finalize<<<cdiv(feat_elems, B), B, 0, stream>>>(T2, x, feat_elems);

    cheb_gemm_bn<HID_C, true><<<dim3(gx, HID_C / 32), B, 0, stream>>>(
        x, T1, T2, W1, b1, g1, beta1, m1, v1, h, N_NODES);

    // ---- layer 2: T0 = h, T1 = L h, T2 = 2 L T1 - h
    zero_f32<<<cdiv(feat_elems, B), B, 0, stream>>>(T1, feat_elems);
    propagate_kernel<<<cdiv(prop_threads, B), B, 0, stream>>>(src, dst, norm, h, T1, E_EDGES);
    zero_f32<<<cdiv(feat_elems, B), B, 0, stream>>>(T2, feat_elems);
    propagate_kernel<<<cdiv(prop_threads, B), B, 0, stream>>>(src, dst, norm, T1, T2, E_EDGES);
    cheb_finalize<<<cdiv(feat_elems, B), B, 0, stream>>>(T2, h, feat_elems);

    cheb_gemm_bn<OUT_C, false><<<dim3(gx, OUT_C / 32), B, 0, stream>>>(
        h, T1, T2, W2, b2, g2, beta2, m2, v2, out, N_NODES);
}